// ARD_82712480187158
// MI455X (gfx1250) — compile-verified
//
#include <hip/hip_runtime.h>
#include <math.h>

// ARD RBF Gram matrix on MI455X (gfx1250, wave32, WMMA).
// out[n,m] = exp(-0.5 * max(x2[n] + y2[m] - 2*<xs_n, ys_m>, 0))
// xs = x * exp(-0.5*logh) (per-feature scale), same for ys.

#define N_ROWS 8192
#define M_ROWS 8192
#define D_DIM  512

#define BLK 128          // output tile per block (BLK x BLK)
#define KB  32           // K-slice staged in LDS
#define SK  (KB + 2)     // padded LDS row stride (floats): conflict-free b64 reads

typedef __attribute__((ext_vector_type(2))) float v2f;   // fp32 WMMA A/B frag (16x4 / 4x16)
typedef __attribute__((ext_vector_type(8))) float v8f;   // fp32 WMMA C/D frag (16x16)

// ---------------- Prologue: scaled row norms (one wave32 per row) ----------------
__global__ __launch_bounds__(256)
void ard_norms_kernel(const float* __restrict__ x,
                      const float* __restrict__ y,
                      const float* __restrict__ logh,
                      float* __restrict__ x2,
                      float* __restrict__ y2) {
  const int lane = threadIdx.x & 31;
  const int wave = threadIdx.x >> 5;
  const long long row = (long long)blockIdx.x * 8 + wave;

  const float* src;
  float* dst;
  if (row < N_ROWS) {
    src = x + row * D_DIM;
    dst = x2 + row;
  } else {
    const long long r = row - N_ROWS;
    src = y + r * D_DIM;
    dst = y2 + r;
  }

  float acc = 0.0f;
  for (int d = lane; d < D_DIM; d += 32) {
    const float v = src[d];
    acc = fmaf(v * v, __expf(-logh[d]), acc);   // x^2 * exp(-logh)
  }
#pragma unroll
  for (int off = 16; off > 0; off >>= 1)
    acc += __shfl_down(acc, off, 32);
  if (lane == 0) *dst = acc;
}

// ---------------- Main: tiled fp32 WMMA GEMM + fused RBF epilogue ----------------
__global__ __launch_bounds__(256)
void ard_gemm_kernel(const float* __restrict__ x,
                     const float* __restrict__ y,
                     const float* __restrict__ logh,
                     const float* __restrict__ x2,
                     const float* __restrict__ y2,
                     float* __restrict__ out) {
  __shared__ __align__(16) float sX[BLK * SK];
  __shared__ __align__(16) float sY[BLK * SK];
  __shared__ float sS[D_DIM];   // inv_sqrt_h = exp(-0.5*logh)

  const int tid  = threadIdx.x;
  const int lane = tid & 31;
  const int wave = tid >> 5;
  const int half = lane >> 4;   // 0: lanes 0-15, 1: lanes 16-31
  const int lrow = lane & 15;

  const int mBlk = blockIdx.y * BLK;   // rows of x
  const int nBlk = blockIdx.x * BLK;   // rows of y (cols of output)

  // 8 waves -> 4x2 wave grid; each wave computes 32(M) x 64(N) = 2x4 WMMA tiles
  const int mW = (wave >> 1) * 32;
  const int nW = (wave & 1) * 64;

  // stage per-feature scales once
  for (int d = tid; d < D_DIM; d += 256)
    sS[d] = __expf(-0.5f * logh[d]);
  __syncthreads();

  v8f acc[2][4];
#pragma unroll
  for (int mt = 0; mt < 2; ++mt)
#pragma unroll
    for (int nt = 0; nt < 4; ++nt)
#pragma unroll
      for (int e = 0; e < 8; ++e)
        acc[mt][nt][e] = 0.0f;

  for (int kb = 0; kb < D_DIM; kb += KB) {
    // ---- stage 128xKB tiles of scaled x and y into LDS (float4 global loads) ----
#pragma unroll
    for (int f = tid; f < BLK * (KB / 4); f += 256) {   // 1024 float4s each
      const int row = f >> 3;          // KB/4 == 8 float4 per row
      const int c4  = (f & 7) * 4;
      const float4 xv = *(const float4*)(x + (size_t)(mBlk + row) * D_DIM + kb + c4);
      const float4 yv = *(const float4*)(y + (size_t)(nBlk + row) * D_DIM + kb + c4);
      const float s0 = sS[kb + c4 + 0];
      const float s1 = sS[kb + c4 + 1];
      const float s2 = sS[kb + c4 + 2];
      const float s3 = sS[kb + c4 + 3];
      float* px = sX + row * SK + c4;
      float* py = sY + row * SK + c4;
      px[0] = xv.x * s0; px[1] = xv.y * s1; px[2] = xv.z * s2; px[3] = xv.w * s3;
      py[0] = yv.x * s0; py[1] = yv.y * s1; py[2] = yv.z * s2; py[3] = yv.w * s3;
    }
    __syncthreads();

    // ---- inner K loop: 4 at a time via V_WMMA_F32_16X16X4_F32 ----
#pragma unroll
    for (int k0 = 0; k0 < KB; k0 += 4) {
      const int kf = k0 + 2 * half;    // even -> 8B-aligned b64 LDS reads
      v2f a[2], b[4];
#pragma unroll
      for (int mt = 0; mt < 2; ++mt)   // A[m][2*half + v], m = lrow
        a[mt] = *(const v2f*)(sX + (mW + mt * 16 + lrow) * SK + kf);
#pragma unroll
      for (int nt = 0; nt < 4; ++nt)   // B[2*half + v][n] = ys[n][k], n = lrow
        b[nt] = *(const v2f*)(sY + (nW + nt * 16 + lrow) * SK + kf);
#pragma unroll
      for (int mt = 0; mt < 2; ++mt)
#pragma unroll
        for (int nt = 0; nt < 4; ++nt)
          acc[mt][nt] = __builtin_amdgcn_wmma_f32_16x16x4_f32(
              false, a[mt], false, b[nt], (short)0, acc[mt][nt], false, false);
    }
    __syncthreads();
  }

  // ---- fused epilogue: sq = x2 + y2 - 2*dot; out = exp(-0.5*max(sq,0)) ----
#pragma unroll
  for (int mt = 0; mt < 2; ++mt) {
#pragma unroll
    for (int nt = 0; nt < 4; ++nt) {
      const int n = nBlk + nW + nt * 16 + lrow;
      const float yn = y2[n];
#pragma unroll
      for (int v = 0; v < 8; ++v) {
        const int m = mBlk + mW + mt * 16 + v + 8 * half;   // C/D layout: M = v + 8*half
        float sq = x2[m] + yn - 2.0f * acc[mt][nt][v];
        sq = fmaxf(sq, 0.0f);
        out[(size_t)m * M_ROWS + n] = __expf(-0.5f * sq);
      }
    }
  }
}

extern "C" void kernel_launch(void* const* d_in, const int* in_sizes, int n_in,
                              void* d_out, int out_size, void* d_ws, size_t ws_size,
                              hipStream_t stream) {
  (void)in_sizes; (void)n_in; (void)out_size; (void)ws_size;
  const float* x    = (const float*)d_in[0];   // [N, D] fp32
  const float* y    = (const float*)d_in[1];   // [M, D] fp32
  const float* logh = (const float*)d_in[2];   // [D]    fp32
  float* out = (float*)d_out;                  // [N, M] fp32

  float* x2 = (float*)d_ws;                    // [N] scaled row norms
  float* y2 = x2 + N_ROWS;                     // [M]

  ard_norms_kernel<<<(N_ROWS + M_ROWS) / 8, 256, 0, stream>>>(x, y, logh, x2, y2);

  dim3 grid(M_ROWS / BLK, N_ROWS / BLK);       // (64, 64)
  ard_gemm_kernel<<<grid, 256, 0, stream>>>(x, y, logh, x2, y2, out);
}